// impactDetect_48309792145899
// MI455X (gfx1250) — compile-verified
//
#include <hip/hip_runtime.h>
#include <hip/hip_bf16.h>

typedef float v2f __attribute__((ext_vector_type(2)));
typedef float v8f __attribute__((ext_vector_type(8)));

#define HDIM 64

// ---------------------------------------------------------------------------
// zero-fill
__global__ void zero_kernel(float* __restrict__ p, long long n) {
  long long i = (long long)blockIdx.x * blockDim.x + threadIdx.x;
  if (i < n) p[i] = 0.0f;
}

// ---------------------------------------------------------------------------
// degree accumulation: deg[dst[e]] += 1
__global__ void degree_kernel(const int* __restrict__ dst, float* __restrict__ deg,
                              long long E) {
  long long e = (long long)blockIdx.x * blockDim.x + threadIdx.x;
  if (e < E) atomicAdd(&deg[dst[e]], 1.0f);
}

// dinv[i] = rsqrt(deg[i] + 1)   (self loop), in place
__global__ void rsqrt_kernel(float* __restrict__ d, int n) {
  int i = blockIdx.x * blockDim.x + threadIdx.x;
  if (i < n) d[i] = rsqrtf(d[i] + 1.0f);
}

// ---------------------------------------------------------------------------
// C[nrows,64] = X[nrows,64] @ W[64,64] via V_WMMA_F32_16X16X4_F32.
// One wave handles a 16-row M-tile, all 4 column tiles, 16 K-steps of depth 4.
// A 16x4 f32 layout: lanes 0-15 -> M=lane, K={0,1}; lanes 16-31 -> K={2,3}.
// B 4x16 f32 layout mirrors A with N = lane%16.
// C/D: 8 VGPRs; lane<16 -> M=r, lane>=16 -> M=r+8; N = lane%16.
__global__ void gemm64_wmma(const float* __restrict__ X,
                            const float* __restrict__ W,
                            float* __restrict__ C, int nrows) {
  const int lane   = threadIdx.x & 31;
  const int wave   = threadIdx.x >> 5;
  const int wavesB = blockDim.x >> 5;
  const int tile   = blockIdx.x * wavesB + wave;       // M-tile (wave-uniform)
  const int m0     = tile * 16;
  if (m0 >= nrows) return;                              // wave-uniform exit

  const int mrow  = lane & 15;
  const int khalf = lane >> 4;                          // 0 or 1

  // preload A fragments for this 16-row stripe
  v2f a[16];
  const float* xrow = X + (size_t)(m0 + mrow) * HDIM;
  #pragma unroll
  for (int ks = 0; ks < 16; ++ks) {
    const int k = ks * 4 + khalf * 2;
    a[ks].x = xrow[k];
    a[ks].y = xrow[k + 1];
  }

  #pragma unroll
  for (int nt = 0; nt < 4; ++nt) {
    const int n0 = nt * 16;
    v8f c = {};
    #pragma unroll
    for (int ks = 0; ks < 16; ++ks) {
      const int k = ks * 4 + khalf * 2;
      v2f b;
      b.x = W[(size_t)k       * HDIM + n0 + mrow];
      b.y = W[(size_t)(k + 1) * HDIM + n0 + mrow];
      c = __builtin_amdgcn_wmma_f32_16x16x4_f32(false, a[ks], false, b,
                                                (short)0, c, false, false);
    }
    #pragma unroll
    for (int r = 0; r < 8; ++r) {
      C[(size_t)(m0 + r + 8 * khalf) * HDIM + n0 + mrow] = c[r];
    }
  }
}

// ---------------------------------------------------------------------------
// edge scatter: agg[dst] += h[src] * dinv[src]*dinv[dst]
// 16 lanes per edge, each handling 4 consecutive features (float4 gather).
__global__ void scatter_kernel(const int* __restrict__ src,
                               const int* __restrict__ dst,
                               const float* __restrict__ h,
                               const float* __restrict__ dinv,
                               float* __restrict__ agg, long long E) {
  long long idx = (long long)blockIdx.x * blockDim.x + threadIdx.x;
  long long e  = idx >> 4;
  int       kq = (int)(idx & 15) << 2;
  if (e >= E) return;
  const int s = src[e];
  const int d = dst[e];
  const float coef = dinv[s] * dinv[d];
  const float4 hv = *(const float4*)(h + (size_t)s * HDIM + kq);
  float* ap = agg + (size_t)d * HDIM + kq;
  atomicAdd(ap + 0, hv.x * coef);
  atomicAdd(ap + 1, hv.y * coef);
  atomicAdd(ap + 2, hv.z * coef);
  atomicAdd(ap + 3, hv.w * coef);
}

// ---------------------------------------------------------------------------
// z = [relu](agg + h * dinv^2 + b)     elementwise over [N,64]
__global__ void combine_kernel(const float* __restrict__ agg,
                               const float* __restrict__ h,
                               const float* __restrict__ dinv,
                               const float* __restrict__ bias,
                               float* __restrict__ out,
                               long long total, int doRelu) {
  long long idx = (long long)blockIdx.x * blockDim.x + threadIdx.x;
  if (idx >= total) return;
  const long long i = idx >> 6;
  const int k = (int)(idx & 63);
  const float di = dinv[i];
  float v = agg[idx] + h[idx] * di * di + bias[k];
  if (doRelu) v = fmaxf(v, 0.0f);
  out[idx] = v;
}

// ---------------------------------------------------------------------------
// three fused [64->1] heads over the real-graph z2 (reads each row once)
__global__ void heads3_kernel(const float* __restrict__ z2,
                              const float* __restrict__ Wy, const float* __restrict__ by,
                              const float* __restrict__ Wp, const float* __restrict__ bp,
                              const float* __restrict__ Wb, const float* __restrict__ bb,
                              float* __restrict__ o_yi, float* __restrict__ o_fp,
                              float* __restrict__ o_tp, int n) {
  int i = blockIdx.x * blockDim.x + threadIdx.x;
  if (i >= n) return;
  const float* row = z2 + (size_t)i * HDIM;
  float sy = 0.0f, sp = 0.0f, sb = 0.0f;
  #pragma unroll
  for (int k = 0; k < HDIM; ++k) {
    const float v = row[k];
    sy = fmaf(v, Wy[k], sy);
    sp = fmaf(v, Wp[k], sp);
    sb = fmaf(v, Wb[k], sb);
  }
  o_yi[i] = fmaxf(sy + by[0], 0.0f);
  o_fp[i] = fmaxf(sp + bp[0], 0.0f);
  o_tp[i] = fmaxf(sb + bb[0], 0.0f);
}

// single head for the fake graph
__global__ void head1_kernel(const float* __restrict__ z2,
                             const float* __restrict__ Wp, const float* __restrict__ bp,
                             float* __restrict__ o, int n) {
  int i = blockIdx.x * blockDim.x + threadIdx.x;
  if (i >= n) return;
  const float* row = z2 + (size_t)i * HDIM;
  float sp = 0.0f;
  #pragma unroll
  for (int k = 0; k < HDIM; ++k) sp = fmaf(row[k], Wp[k], sp);
  o[i] = fmaxf(sp + bp[0], 0.0f);
}

// ---------------------------------------------------------------------------
static inline int cdiv_ll(long long a, int b) { return (int)((a + b - 1) / b); }

extern "C" void kernel_launch(void* const* d_in, const int* in_sizes, int n_in,
                              void* d_out, int out_size, void* d_ws, size_t ws_size,
                              hipStream_t stream) {
  const float* x   = (const float*)d_in[0];
  const int*   ei  = (const int*)d_in[1];
  const float* xf  = (const float*)d_in[2];
  const int*   eif = (const int*)d_in[3];
  const float* W1  = (const float*)d_in[4];
  const float* b1  = (const float*)d_in[5];
  const float* W2  = (const float*)d_in[6];
  const float* b2  = (const float*)d_in[7];
  const float* Wy  = (const float*)d_in[8];
  const float* by  = (const float*)d_in[9];
  const float* Wp  = (const float*)d_in[10];
  const float* bp  = (const float*)d_in[11];
  const float* Wb  = (const float*)d_in[12];
  const float* bb  = (const float*)d_in[13];
  float* out = (float*)d_out;

  const int       N  = in_sizes[0] / HDIM;        // 100000
  const long long E  = (long long)in_sizes[1] / 2; // 1600000
  const long long NH = (long long)N * HDIM;

  float* ws   = (float*)d_ws;
  float* dinv = ws;            // N
  float* A    = ws + N;        // N*64  (h buffer)
  float* B    = A + NH;        // N*64  (agg / z buffer)

  const int TB = 256;
  const int gemmTiles  = (N + 15) / 16;
  const int gemmBlocks = (gemmTiles + (TB / 32) - 1) / (TB / 32);
  const int scatBlocks = cdiv_ll(E * 16, TB);
  const int elemBlocks = cdiv_ll(NH, TB);
  const int nodeBlocks = cdiv_ll(N, TB);
  const int edgeBlocks = cdiv_ll(E, TB);

  // -------- fake graph first (buffers recycled for real graph afterwards) ----
  {
    const int* src = eif;
    const int* dst = eif + E;
    zero_kernel<<<nodeBlocks, TB, 0, stream>>>(dinv, N);
    degree_kernel<<<edgeBlocks, TB, 0, stream>>>(dst, dinv, E);
    rsqrt_kernel<<<nodeBlocks, TB, 0, stream>>>(dinv, N);

    gemm64_wmma<<<gemmBlocks, TB, 0, stream>>>(xf, W1, A, N);      // h1f
    zero_kernel<<<elemBlocks, TB, 0, stream>>>(B, NH);
    scatter_kernel<<<scatBlocks, TB, 0, stream>>>(src, dst, A, dinv, B, E);
    combine_kernel<<<elemBlocks, TB, 0, stream>>>(B, A, dinv, b1, B, NH, 1); // z1f

    gemm64_wmma<<<gemmBlocks, TB, 0, stream>>>(B, W2, A, N);       // h2f
    zero_kernel<<<elemBlocks, TB, 0, stream>>>(B, NH);
    scatter_kernel<<<scatBlocks, TB, 0, stream>>>(src, dst, A, dinv, B, E);
    combine_kernel<<<elemBlocks, TB, 0, stream>>>(B, A, dinv, b2, B, NH, 0); // z2f

    head1_kernel<<<nodeBlocks, TB, 0, stream>>>(B, Wp, bp, out + 2LL * N, N); // fprob_f
  }

  // -------- real graph ------------------------------------------------------
  {
    const int* src = ei;
    const int* dst = ei + E;
    zero_kernel<<<nodeBlocks, TB, 0, stream>>>(dinv, N);
    degree_kernel<<<edgeBlocks, TB, 0, stream>>>(dst, dinv, E);
    rsqrt_kernel<<<nodeBlocks, TB, 0, stream>>>(dinv, N);

    gemm64_wmma<<<gemmBlocks, TB, 0, stream>>>(x, W1, A, N);       // h1
    zero_kernel<<<elemBlocks, TB, 0, stream>>>(B, NH);
    scatter_kernel<<<scatBlocks, TB, 0, stream>>>(src, dst, A, dinv, B, E);
    combine_kernel<<<elemBlocks, TB, 0, stream>>>(B, A, dinv, b1, B, NH, 1); // z1

    gemm64_wmma<<<gemmBlocks, TB, 0, stream>>>(B, W2, A, N);       // h2
    zero_kernel<<<elemBlocks, TB, 0, stream>>>(B, NH);
    scatter_kernel<<<scatBlocks, TB, 0, stream>>>(src, dst, A, dinv, B, E);
    combine_kernel<<<elemBlocks, TB, 0, stream>>>(B, A, dinv, b2, B, NH, 0); // z2

    heads3_kernel<<<nodeBlocks, TB, 0, stream>>>(B, Wy, by, Wp, bp, Wb, bb,
                                                 out, out + N, out + 3LL * N, N);
  }
}